// LinearAttention_46359876993671
// MI455X (gfx1250) — compile-verified
//
#include <hip/hip_runtime.h>
#include <stdint.h>

typedef __attribute__((ext_vector_type(2))) float v2f;
typedef __attribute__((ext_vector_type(8))) float v8f;

namespace {
constexpr int T  = 4096;     // sequence length
constexpr int Dh = 64;       // head dim
constexpr int BH = 16;       // batch*heads = 2*8
constexpr int CH = 64;       // chunk length
constexpr int NC = T / CH;   // chunks per head
constexpr int LD = 72;       // padded LDS row stride (floats; 288B, 16B-aligned)
constexpr int NTHREADS = 128; // 4 x wave32
}

// D(16x16,f32) = A(16x4,f32) @ B(4x16,f32) + C   — full-precision WMMA path.
__device__ __forceinline__ v8f wmma_f32(v2f a, v2f b, v8f c) {
  return __builtin_amdgcn_wmma_f32_16x16x4_f32(false, a, false, b,
                                               (short)0, c, false, false);
}

// Load a [CH][Dh] fp32 tile from global, apply RoPE (theta=1e4, half=32),
// store to LDS with row stride LD.
__device__ __forceinline__ void load_rope_tile(const float* __restrict__ src,
                                               float* __restrict__ dst,
                                               int tBase, int tid) {
  for (int p = tid; p < CH * 32; p += NTHREADS) {
    const int t = p >> 5;
    const int j = p & 31;
    const float x1 = src[t * Dh + j];
    const float x2 = src[t * Dh + j + 32];
    // inv_freq = 10000^(-j/32) = exp(-j * ln(1e4)/32)
    const float freq = __expf(-0.28782313662425572f * (float)j);
    const float ang  = (float)(tBase + t) * freq;
    float s, c;
    sincosf(ang, &s, &c);
    dst[t * LD + j]      = x1 * c - x2 * s;
    dst[t * LD + j + 32] = x1 * s + x2 * c;
  }
}

// Asynchronously copy a [CH][Dh] fp32 tile (row-major, contiguous) from global
// memory directly into LDS (row stride LD), 16 bytes per lane per issue.
// Uses the CDNA5 ASYNCcnt path: no VGPR round-trip, overlaps with VALU work.
__device__ __forceinline__ void async_copy_tile(const float* __restrict__ src,
                                                float* __restrict__ dstLds,
                                                int tid) {
#pragma unroll
  for (int i = 0; i < (CH * Dh) / (4 * NTHREADS); ++i) {  // 8 issues/thread
    const int q   = tid + i * NTHREADS;   // 16-byte chunk index
    const int row = q >> 4;               // tile row (16 chunks per row)
    const int c4  = q & 15;               // chunk within row
    const unsigned ldsAddr = (unsigned)(uintptr_t)(dstLds + row * LD + c4 * 4);
    const unsigned gOff    = (unsigned)((row * Dh + c4 * 4) * sizeof(float));
    asm volatile("global_load_async_to_lds_b128 %0, %1, %2"
                 :: "v"(ldsAddr), "v"(gOff), "s"(src)
                 : "memory");
  }
}

__device__ __forceinline__ void wait_async_and_barrier() {
  asm volatile("s_wait_asynccnt 0x0" ::: "memory");
  __syncthreads();
}

// Kernel 1: per (head, chunk) compute S_c = rope(K_c)^T @ V_c  (64x64).
__global__ __launch_bounds__(NTHREADS) void rope_chunk_kv_kernel(
    const float* __restrict__ K, const float* __restrict__ V,
    float* __restrict__ S) {
  extern __shared__ float smem[];
  float* sK = smem;               // roped K chunk [CH][LD]
  float* sV = smem + CH * LD;     // V chunk       [CH][LD]

  const int bh  = blockIdx.y;
  const int c   = blockIdx.x;
  const int tid = threadIdx.x;

  const size_t rowBase = ((size_t)bh * T + (size_t)c * CH) * Dh;
  async_copy_tile(V + rowBase, sV, tid);       // async V copy overlaps RoPE
  load_rope_tile(K + rowBase, sK, c * CH, tid);
  wait_async_and_barrier();

  const int wave = tid >> 5;
  const int lane = tid & 31;
  const int l16  = lane & 15;
  const int hi   = lane >> 4;

  v8f acc[4];
  acc[0] = {}; acc[1] = {}; acc[2] = {}; acc[3] = {};

  const int m = wave * 16 + l16;          // output row = d_k index
  for (int k0 = 0; k0 < CH; k0 += 4) {    // reduce over time within chunk
    const int kk = k0 + hi * 2;
    v2f a;                                // A[m][kk] = Kr^T[m][kk] = Kr[kk][m]
    a.x = sK[kk * LD + m];
    a.y = sK[(kk + 1) * LD + m];
#pragma unroll
    for (int ni = 0; ni < 4; ++ni) {
      const int n = ni * 16 + l16;        // output col = d_v index
      v2f b;                              // B[kk][n] = V[kk][n]
      b.x = sV[kk * LD + n];
      b.y = sV[(kk + 1) * LD + n];
      acc[ni] = wmma_f32(a, b, acc[ni]);
    }
  }

  float* Sb = S + ((size_t)bh * NC + c) * (Dh * Dh);
#pragma unroll
  for (int ni = 0; ni < 4; ++ni) {
#pragma unroll
    for (int i = 0; i < 8; ++i) {
      const int row = wave * 16 + i + hi * 8;
      const int col = ni * 16 + l16;
      Sb[row * Dh + col] = acc[ni][i];
    }
  }
}

// Kernel 2: per head, in-place *exclusive* prefix sum of the 64x64 chunk
// states over the chunk axis. One block per head; each thread owns 16 lanes
// of the 4096-element state.
__global__ __launch_bounds__(256) void chunk_scan_kernel(float* __restrict__ S) {
  const int bh  = blockIdx.x;
  const int tid = threadIdx.x;
  float run[16];
#pragma unroll
  for (int e = 0; e < 16; ++e) run[e] = 0.f;
  for (int c = 0; c < NC; ++c) {
    float* p = S + ((size_t)bh * NC + c) * (Dh * Dh);
#pragma unroll
    for (int e = 0; e < 16; ++e) {
      const int idx = e * 256 + tid;
      const float v = p[idx];
      p[idx] = run[e];
      run[e] += v;
    }
  }
}

// Kernel 3: per (head, chunk):
//   out_c = tril_strict(rope(Q_c) @ rope(K_c)^T) @ V_c + rope(Q_c) @ P_c
__global__ __launch_bounds__(NTHREADS) void attn_out_kernel(
    const float* __restrict__ Q, const float* __restrict__ K,
    const float* __restrict__ V, const float* __restrict__ P,
    float* __restrict__ O) {
  extern __shared__ float smem[];
  float* sQ  = smem;                   // roped Q  [CH][LD]
  float* sKP = smem + 1 * CH * LD;     // roped K, later prefix state P
  float* sV  = smem + 2 * CH * LD;     // V        [CH][LD]
  float* sS  = smem + 3 * CH * LD;     // masked scores [CH][LD]

  const int bh  = blockIdx.y;
  const int c   = blockIdx.x;
  const int tid = threadIdx.x;

  const size_t rowBase = ((size_t)bh * T + (size_t)c * CH) * Dh;
  async_copy_tile(V + rowBase, sV, tid);       // async V copy overlaps RoPE
  load_rope_tile(Q + rowBase, sQ,  c * CH, tid);
  load_rope_tile(K + rowBase, sKP, c * CH, tid);
  wait_async_and_barrier();

  const int wave = tid >> 5;
  const int lane = tid & 31;
  const int l16  = lane & 15;
  const int hi   = lane >> 4;
  const int m    = wave * 16 + l16;    // this wave's query-row for A frags

  // ---- Phase A: scores = tril_strict(Qr @ Kr^T), written to LDS ----
  {
    v8f acc[4];
    acc[0] = {}; acc[1] = {}; acc[2] = {}; acc[3] = {};
    for (int k0 = 0; k0 < Dh; k0 += 4) {   // reduce over feature dim
      const int kk = k0 + hi * 2;
      v2f a;                               // A[m][kk] = Qr[m][kk]
      a.x = sQ[m * LD + kk];
      a.y = sQ[m * LD + kk + 1];
#pragma unroll
      for (int ni = 0; ni < 4; ++ni) {
        const int n = ni * 16 + l16;       // key time index
        v2f b;                             // B[kk][n] = Kr^T[kk][n] = Kr[n][kk]
        b.x = sKP[n * LD + kk];
        b.y = sKP[n * LD + kk + 1];
        acc[ni] = wmma_f32(a, b, acc[ni]);
      }
    }
#pragma unroll
    for (int ni = 0; ni < 4; ++ni) {
#pragma unroll
      for (int i = 0; i < 8; ++i) {
        const int qr = wave * 16 + i + hi * 8;
        const int kc = ni * 16 + l16;
        sS[qr * LD + kc] = (qr > kc) ? acc[ni][i] : 0.f;  // strict causal mask
      }
    }
  }
  __syncthreads();   // all waves done reading sKP (K) and writing sS

  // Asynchronously reload sKP with inter-chunk prefix state P_c (d_k x d_v).
  async_copy_tile(P + ((size_t)bh * NC + c) * (Dh * Dh), sKP, tid);
  wait_async_and_barrier();

  // ---- Phase B: out = scores @ V + Qr @ P ----
#pragma unroll
  for (int ni = 0; ni < 4; ++ni) {
    v8f acc = {};
    const int n = ni * 16 + l16;           // d_v output column
    for (int k0 = 0; k0 < CH; k0 += 4) {   // reduce over key time
      const int kk = k0 + hi * 2;
      v2f a, b;
      a.x = sS[m * LD + kk];
      a.y = sS[m * LD + kk + 1];
      b.x = sV[kk * LD + n];
      b.y = sV[(kk + 1) * LD + n];
      acc = wmma_f32(a, b, acc);
    }
    for (int k0 = 0; k0 < Dh; k0 += 4) {   // reduce over d_k via prefix state
      const int kk = k0 + hi * 2;
      v2f a, b;
      a.x = sQ[m * LD + kk];
      a.y = sQ[m * LD + kk + 1];
      b.x = sKP[kk * LD + n];
      b.y = sKP[(kk + 1) * LD + n];
      acc = wmma_f32(a, b, acc);
    }
#pragma unroll
    for (int i = 0; i < 8; ++i) {
      const int qr = wave * 16 + i + hi * 8;
      O[rowBase + (size_t)qr * Dh + n] = acc[i];
    }
  }
}

extern "C" void kernel_launch(void* const* d_in, const int* in_sizes, int n_in,
                              void* d_out, int out_size, void* d_ws, size_t ws_size,
                              hipStream_t stream) {
  (void)in_sizes; (void)n_in; (void)out_size; (void)ws_size;
  const float* Q = (const float*)d_in[0];
  const float* K = (const float*)d_in[1];
  const float* V = (const float*)d_in[2];
  float* O = (float*)d_out;
  float* S = (float*)d_ws;   // BH*NC*64*64 floats = 16 MiB of chunk states

  dim3 grid(NC, BH);
  const size_t smem2 = (size_t)2 * CH * LD * sizeof(float); // 36 KiB
  const size_t smem4 = (size_t)4 * CH * LD * sizeof(float); // 72 KiB

  rope_chunk_kv_kernel<<<grid, NTHREADS, smem2, stream>>>(K, V, S);
  chunk_scan_kernel<<<dim3(BH), 256, 0, stream>>>(S);
  attn_out_kernel<<<grid, NTHREADS, smem4, stream>>>(Q, K, V, S, O);
}